// SU2Attention_89464168775881
// MI455X (gfx1250) — compile-verified
//
#include <hip/hip_runtime.h>
#include <hip/hip_bf16.h>
#include <math.h>

// SU(2) spinor attention, MI455X (gfx1250, wave32).
// Per head: V (2048x4 real view of complex 2-spinors), logits = Vn Vn^T (Vn row-normalized),
// softmax(logits/sqrt(2)) over keys, out = A @ V (unnormalized). Output complex64 layout
// (re0,im0,re1,im1) == our 4-vector order, so each token's output is one float4 store.
//
// Score tiles via V_WMMA_F32_16X16X4_F32 (contraction depth is exactly K=4 -> one WMMA
// per 16x16 tile). We compute the TRANSPOSED score tile (A = key chunk 16x4, B = Q^T 4x16)
// so each lane owns one query column: online softmax is lane-local + one xor-16 exchange.
//
// Constant folding: staged inverse norm carries c = sqrt(SCALE * log2(e)), so the WMMA
// output is already "log2-domain scaled logits" and every exponential is a bare
// v_exp_f32 (exp2) with no pre-multiplies. Softmax is base-invariant, result identical.

typedef __attribute__((ext_vector_type(2))) float v2f;
typedef __attribute__((ext_vector_type(4))) float v4f;
typedef __attribute__((ext_vector_type(8))) float v8f;

#define SEQ   2048
#define HEADS 8
#define SU2_SCALE 0.70710678118654752440f   // 1/sqrt(2)
#define LOG2E     1.44269504088896340736f

__global__ __launch_bounds__(512)
void su2_attn_kernel(const float* __restrict__ sreal,
                     const float* __restrict__ simag,
                     float* __restrict__ out) {
  // LDS: unnormalized 4-vectors S (2048x4 = 32KB) + folded inverse norms (2048 = 8KB)
  __shared__ float ldsS[SEQ * 4];
  __shared__ float ldsInv[SEQ];

  const int bid  = blockIdx.x;
  const int h    = bid & (HEADS - 1);   // head
  const int g    = bid >> 3;            // query group (0..7)
  const int tid  = threadIdx.x;
  const int lane = tid & 31;
  const int wave = tid >> 5;            // 0..15
  const int half = lane >> 4;           // 0 or 1 (half-wave)
  const int l16  = lane & 15;

  // c^2 * (v_i . v_j) = logits * SCALE * log2(e)  ==> exp2() gives softmax numerator
  const float cfold = __builtin_sqrtf(SU2_SCALE * LOG2E);

  // ---- Stage this head's spinors into LDS (4 rows per thread) ----
  #pragma unroll
  for (int r = 0; r < 4; ++r) {
    const int j = (tid << 2) + r;
    const v2f re = *(const v2f*)(sreal + j * (HEADS * 2) + h * 2);  // (re0, re1)
    const v2f im = *(const v2f*)(simag + j * (HEADS * 2) + h * 2);  // (im0, im1)
    const float s0 = re.x, s1 = im.x, s2 = re.y, s3 = im.y;         // (re0,im0,re1,im1)
    ldsS[j * 4 + 0] = s0;
    ldsS[j * 4 + 1] = s1;
    ldsS[j * 4 + 2] = s2;
    ldsS[j * 4 + 3] = s3;
    ldsInv[j] = rsqrtf(s0 * s0 + s1 * s1 + s2 * s2 + s3 * s3) * cfold;
  }
  __syncthreads();

  // ---- Each wave owns one 16-query tile ----
  const int qbase = (g * 16 + wave) * 16;

  // B matrix (Q^T, 4x16): lane holds B[K = 2*half + v][N = l16] = c*Vn[qbase+l16][2*half+v]
  v2f bq;
  {
    const v2f sq = *(const v2f*)&ldsS[(qbase + l16) * 4 + 2 * half];
    const float iq = ldsInv[qbase + l16];
    bq.x = sq.x * iq;
    bq.y = sq.y * iq;
  }

  float row_m = -INFINITY;  // running max (log2 domain; mirrored in both half-lanes)
  float row_l = 0.0f;       // running denom
  float o0 = 0.f, o1 = 0.f, o2 = 0.f, o3 = 0.f;  // partial output (this half-lane's j's)

  for (int c = 0; c < SEQ / 16; ++c) {
    const int jb = c * 16;

    // A matrix (key chunk, 16x4): lane holds A[M=l16][K=2*half+v] = c*Vn[jb+l16][2*half+v]
    v2f a;
    {
      const v2f sa = *(const v2f*)&ldsS[(jb + l16) * 4 + 2 * half];
      const float ia = ldsInv[jb + l16];
      a.x = sa.x * ia;
      a.y = sa.y * ia;
    }

    // St[M=j-jb][N=i-qbase]: one f32 WMMA gives the whole 16x16 scaled-logit tile (K=4).
    v8f st = {0.f, 0.f, 0.f, 0.f, 0.f, 0.f, 0.f, 0.f};
    st = __builtin_amdgcn_wmma_f32_16x16x4_f32(
        /*neg_a=*/false, a, /*neg_b=*/false, bq,
        /*c_mod=*/(short)0, st, /*reuse_a=*/false, /*reuse_b=*/false);

    // Lane L, VGPR r holds St[j = jb + 8*half + r][i = qbase + l16] (log2-scaled).
    float cmax = -INFINITY;
    #pragma unroll
    for (int r = 0; r < 8; ++r) cmax = fmaxf(cmax, st[r]);
    cmax = fmaxf(cmax, __shfl_xor(cmax, 16, 32));   // join the two half-lanes of query i

    const float m_new = fmaxf(row_m, cmax);
    const float alpha = __builtin_amdgcn_exp2f(row_m - m_new);  // first chunk: exp2(-inf)=0
    float p[8];
    float lsum = 0.f;
    #pragma unroll
    for (int r = 0; r < 8; ++r) {
      p[r] = __builtin_amdgcn_exp2f(st[r] - m_new);
      lsum += p[r];
    }
    row_l = row_l * alpha + lsum + __shfl_xor(lsum, 16, 32);
    row_m = m_new;

    // P @ V for this chunk: 4-wide output, lane-local FMAs against LDS rows (broadcast reads)
    o0 *= alpha; o1 *= alpha; o2 *= alpha; o3 *= alpha;
    #pragma unroll
    for (int r = 0; r < 8; ++r) {
      const v4f sv = *(const v4f*)&ldsS[(jb + half * 8 + r) * 4];
      o0 = fmaf(p[r], sv.x, o0);
      o1 = fmaf(p[r], sv.y, o1);
      o2 = fmaf(p[r], sv.z, o2);
      o3 = fmaf(p[r], sv.w, o3);
    }
  }

  // Combine the two half-lane partials and write: out[i,h] float4 at i*32 + h*4
  o0 += __shfl_xor(o0, 16, 32);
  o1 += __shfl_xor(o1, 16, 32);
  o2 += __shfl_xor(o2, 16, 32);
  o3 += __shfl_xor(o3, 16, 32);
  const float invl = 1.0f / row_l;
  if (half == 0) {
    const int i = qbase + l16;
    v4f o = {o0 * invl, o1 * invl, o2 * invl, o3 * invl};
    *(v4f*)&out[i * (HEADS * 4) + h * 4] = o;
  }
}

extern "C" void kernel_launch(void* const* d_in, const int* in_sizes, int n_in,
                              void* d_out, int out_size, void* d_ws, size_t ws_size,
                              hipStream_t stream) {
  (void)in_sizes; (void)n_in; (void)d_ws; (void)ws_size; (void)out_size;
  const float* sreal = (const float*)d_in[0];
  const float* simag = (const float*)d_in[1];
  float* out = (float*)d_out;  // complex64 flattened as interleaved float pairs

  // 8 heads x 8 query groups = 64 blocks; 512 threads = 16 waves, one 16-query tile each.
  su2_attn_kernel<<<dim3(HEADS * 8), dim3(512), 0, stream>>>(sreal, simag, out);
}